// NeuronPopulation_26336739459345
// MI455X (gfx1250) — compile-verified
//
#include <hip/hip_runtime.h>
#include <hip/hip_bf16.h>
#include <stdint.h>

#define NN        4096
#define NBATCH    16384
#define KSEL      409u
#define THREADS   256
#define NCHUNK    4          // float4 chunks per thread: 4 * 256 * 4 = 4096 elems

// Native 4-float vector for nontemporal stores (HIP float4 is a class type,
// which __builtin_nontemporal_store rejects).
typedef float v4f __attribute__((ext_vector_type(4)));

// Monotonic (ascending) unsigned transform of float bits.
__device__ __forceinline__ unsigned orderKey(float f) {
  unsigned u = __float_as_uint(f);
  return (u & 0x80000000u) ? ~u : (u | 0x80000000u);
}

__device__ __forceinline__ float gelu_exact(float v) {
  return 0.5f * v * (1.0f + erff(v * 0.70710678118654752440f));
}

__global__ __launch_bounds__(THREADS)
void neuronpop_fused_kernel(const float* __restrict__ x,
                            const float* __restrict__ mod,
                            const float* __restrict__ gamma,
                            const float* __restrict__ beta,
                            float* __restrict__ out,
                            float* __restrict__ maskOut)
{
  __shared__ __align__(16) float ldsX[NN];   // x row, later GELU output row
  __shared__ __align__(16) float ldsM[NN];   // modulation row
  __shared__ unsigned hist[256];
  __shared__ float wsum[8], wsq[8];
  __shared__ float sMean, sRsig;
  __shared__ unsigned sPrefix, sRemK;

  const int tid = threadIdx.x;
  const int row = blockIdx.x;

  const unsigned long long xrow =
      (unsigned long long)(const void*)(x + (size_t)row * NN);
  const unsigned long long mrow =
      (unsigned long long)(const void*)(mod + (size_t)row * NN);

  // LDS byte offsets of the staging buffers (low 32 bits of the generic
  // address of a __shared__ object are the LDS offset on CDNA5).
  const unsigned ldsXBase = (unsigned)(uintptr_t)(void*)&ldsX[0];
  const unsigned ldsMBase = (unsigned)(uintptr_t)(void*)&ldsM[0];

  // -------- Phase A: async global -> LDS copies (ASYNCcnt data-mover path) --
  // GVS mode: mem = SGPR64 base + VGPR32 byte offset; LDS dest addr in VGPR.
#pragma unroll
  for (int i = 0; i < NCHUNK; ++i) {
    unsigned off  = (unsigned)((i * THREADS + tid) * 16);
    unsigned ldsA = ldsXBase + off;
    unsigned ldsB = ldsMBase + off;
    asm volatile("global_load_async_to_lds_b128 %0, %1, %2"
                 :: "v"(ldsA), "v"(off), "s"(xrow) : "memory");
    asm volatile("global_load_async_to_lds_b128 %0, %1, %2"
                 :: "v"(ldsB), "v"(off), "s"(mrow) : "memory");
  }
  asm volatile("s_wait_asynccnt 0" ::: "memory");
  __syncthreads();

  // -------- Phase B: p = x * mod, accumulate row moments ---------------------
  float4* ldsX4 = (float4*)ldsX;
  float4* ldsM4 = (float4*)ldsM;

  float s1 = 0.0f, s2 = 0.0f;
  float4 pv[NCHUNK];
#pragma unroll
  for (int i = 0; i < NCHUNK; ++i) {
    int c = i * THREADS + tid;
    float4 xv = ldsX4[c];
    float4 mv = ldsM4[c];
    float4 p;
    p.x = xv.x * mv.x; p.y = xv.y * mv.y;
    p.z = xv.z * mv.z; p.w = xv.w * mv.w;
    pv[i] = p;
    s1 += p.x + p.y + p.z + p.w;
    s2 += p.x * p.x + p.y * p.y + p.z * p.z + p.w * p.w;
  }
  // wave32 reduction
#pragma unroll
  for (int o = 16; o > 0; o >>= 1) {
    s1 += __shfl_xor(s1, o, 32);
    s2 += __shfl_xor(s2, o, 32);
  }
  const int wave = tid >> 5;
  if ((tid & 31) == 0) { wsum[wave] = s1; wsq[wave] = s2; }
  __syncthreads();
  if (tid == 0) {
    float a = 0.0f, b = 0.0f;
#pragma unroll
    for (int w = 0; w < 8; ++w) { a += wsum[w]; b += wsq[w]; }
    float mean = a * (1.0f / NN);
    float var  = b * (1.0f / NN) - mean * mean;
    sMean = mean;
    sRsig = rsqrtf(var + 1e-5f);
  }
  __syncthreads();
  const float mean = sMean;
  const float rsig = sRsig;

  // -------- Phase C: normalize + affine + exact GELU, result -> ldsX ---------
  const float4* g4 = (const float4*)gamma;
  const float4* b4 = (const float4*)beta;
#pragma unroll
  for (int i = 0; i < NCHUNK; ++i) {
    int c = i * THREADS + tid;
    float4 gm = g4[c];
    float4 bt = b4[c];
    float4 p  = pv[i];
    float4 g;
    g.x = gelu_exact((p.x - mean) * rsig * gm.x + bt.x);
    g.y = gelu_exact((p.y - mean) * rsig * gm.y + bt.y);
    g.z = gelu_exact((p.z - mean) * rsig * gm.z + bt.z);
    g.w = gelu_exact((p.w - mean) * rsig * gm.w + bt.w);
    ldsX4[c] = g;
  }
  __syncthreads();

  // -------- Phase D: exact K-th largest via MSB-first radix select -----------
  unsigned prefix = 0u, remk = KSEL;
#pragma unroll 1
  for (int pass = 3; pass >= 0; --pass) {
    if (tid < 256) hist[tid] = 0u;
    __syncthreads();

    const unsigned shift  = (unsigned)pass * 8u;
    const unsigned himask = (pass == 3) ? 0u : (0xFFFFFFFFu << (shift + 8u));
#pragma unroll
    for (int i = 0; i < NCHUNK; ++i) {
      int c = i * THREADS + tid;
      float4 g = ldsX4[c];
      unsigned k0 = orderKey(g.x), k1 = orderKey(g.y);
      unsigned k2 = orderKey(g.z), k3 = orderKey(g.w);
      if ((k0 & himask) == prefix) atomicAdd(&hist[(k0 >> shift) & 255u], 1u);
      if ((k1 & himask) == prefix) atomicAdd(&hist[(k1 >> shift) & 255u], 1u);
      if ((k2 & himask) == prefix) atomicAdd(&hist[(k2 >> shift) & 255u], 1u);
      if ((k3 & himask) == prefix) atomicAdd(&hist[(k3 >> shift) & 255u], 1u);
    }
    __syncthreads();

    if (tid == 0) {
      unsigned cum = 0u;
      for (int b = 255; b >= 0; --b) {
        unsigned cb = hist[b];
        if (cum + cb >= remk) {
          sPrefix = prefix | ((unsigned)b << shift);
          sRemK   = remk - cum;
          break;
        }
        cum += cb;
      }
    }
    __syncthreads();
    prefix = sPrefix;
    remk   = sRemK;
    __syncthreads();
  }
  // prefix == exact orderKey of the K-th largest element in this row.

  // -------- Phase E: masked writeback, non-temporal b128 streams -------------
  // Outputs (512 MB) are write-once / never re-read: TH_STORE_NT keeps them
  // from churning the 192 MB L2, preserving it for gamma/beta + input streams.
  v4f* out4 = (v4f*)out;
  v4f* msk4 = (v4f*)maskOut;
  const size_t rowc = (size_t)row * (NN / 4);
#pragma unroll
  for (int i = 0; i < NCHUNK; ++i) {
    int c = i * THREADS + tid;
    float4 g = ldsX4[c];
    v4f m, o;
    m.x = (orderKey(g.x) >= prefix) ? 1.0f : 0.0f;
    m.y = (orderKey(g.y) >= prefix) ? 1.0f : 0.0f;
    m.z = (orderKey(g.z) >= prefix) ? 1.0f : 0.0f;
    m.w = (orderKey(g.w) >= prefix) ? 1.0f : 0.0f;
    o.x = g.x * m.x; o.y = g.y * m.y; o.z = g.z * m.z; o.w = g.w * m.w;
    __builtin_nontemporal_store(o, &out4[rowc + c]);
    __builtin_nontemporal_store(m, &msk4[rowc + c]);
  }
}

extern "C" void kernel_launch(void* const* d_in, const int* in_sizes, int n_in,
                              void* d_out, int out_size, void* d_ws, size_t ws_size,
                              hipStream_t stream) {
  const float* x     = (const float*)d_in[0];
  const float* mod   = (const float*)d_in[1];
  const float* gamma = (const float*)d_in[2];
  const float* beta  = (const float*)d_in[3];
  float* out  = (float*)d_out;
  float* mask = out + (size_t)NBATCH * NN;   // tuple: (out, mask) flat-concatenated

  neuronpop_fused_kernel<<<NBATCH, THREADS, 0, stream>>>(x, mod, gamma, beta,
                                                         out, mask);
}